// RumorDetectionModel_76450417868973
// MI455X (gfx1250) — compile-verified
//
#include <hip/hip_runtime.h>
#include <hip/hip_bf16.h>

#define N_NODES 50000
#define N_EDGES 600000
#define DF 128
#define NG 512
#define NC 2

typedef float v2f __attribute__((ext_vector_type(2)));
typedef float v8f __attribute__((ext_vector_type(8)));

__device__ __forceinline__ v8f wmma_f32(v2f a, v2f b, v8f c) {
  // D = A(16x4 f32) * B(4x16 f32) + C(16x16 f32)
  return __builtin_amdgcn_wmma_f32_16x16x4_f32(
      /*neg_a=*/false, a, /*neg_b=*/false, b,
      /*c_mod=*/(short)0, c, /*reuse_a=*/false, /*reuse_b=*/false);
}

// ---------------------------------------------------------------------------
// Pack B[K x 128] row-major into fragment-native pairs:
// P[p*128 + n] = (B[2p][n], B[2p+1][n])  -> one b64 load per B fragment.
__global__ void pack_pairs_k(const float* __restrict__ B, float2* __restrict__ P,
                             int Kpairs) {
  int i = blockIdx.x * blockDim.x + threadIdx.x;
  if (i >= Kpairs * DF) return;
  int p = i >> 7;
  int n = i & 127;
  float2 v;
  v.x = B[(2 * p) * DF + n];
  v.y = B[(2 * p + 1) * DF + n];
  P[i] = v;
}

// init: deg = 1 (self loop), pooled = 0, counts = 0
__global__ void init_misc_k(float* __restrict__ deg, float* __restrict__ pooled,
                            float* __restrict__ counts) {
  int i = blockIdx.x * blockDim.x + threadIdx.x;
  if (i < N_NODES) deg[i] = 1.0f;
  if (i < NG * DF) pooled[i] = 0.0f;
  if (i < NG) counts[i] = 0.0f;
}

// deg[dst[e]] += 1
__global__ void deg_edges_k(const int* __restrict__ dst, float* __restrict__ deg) {
  int e = blockIdx.x * blockDim.x + threadIdx.x;
  if (e >= N_EDGES) return;
  atomicAdd(&deg[dst[e]], 1.0f);
}

// dinv = rsqrt(deg) in place; counts[batch[i]] += 1
__global__ void dinv_counts_k(float* __restrict__ deg, const int* __restrict__ batch,
                              float* __restrict__ counts) {
  int i = blockIdx.x * blockDim.x + threadIdx.x;
  if (i >= N_NODES) return;
  deg[i] = rsqrtf(deg[i]);
  atomicAdd(&counts[batch[i]], 1.0f);
}

// ---------------------------------------------------------------------------
// C[M x 128] = A[M x 128] @ B[128 x 128]  (MODE 1: + bias, ReLU)
// B passed in packed-pair layout (see pack_pairs_k). One wave = 16 rows x 128
// cols; K loop in steps of 4 via v_wmma_f32_16x16x4_f32.
template <int MODE>
__global__ __launch_bounds__(256) void gemm128_k(const float* __restrict__ A,
                                                 const float* __restrict__ Bp,
                                                 const float* __restrict__ bias,
                                                 float* __restrict__ C, int M) {
  int wave = (blockIdx.x * 256 + (int)threadIdx.x) >> 5;
  int lane = threadIdx.x & 31;
  int m0 = wave * 16;
  if (m0 >= M) return;
  int half = lane >> 4;   // 0: K=k,k+1 ; 1: K=k+2,k+3
  int lr = lane & 15;     // A: row within tile ; B: col within tile
  const float* ap = A + (m0 + lr) * DF + 2 * half;
  v8f z = {};
  v8f acc[8];
#pragma unroll
  for (int t = 0; t < 8; ++t) acc[t] = z;

  for (int k = 0; k < DF; k += 4) {
    v2f a = *(const v2f*)(ap + k);
    // pair index for this half: (k + 2*half)/2 = k/2 + half
    const float* bp = Bp + (((k >> 1) + half) * DF + lr) * 2;
#pragma unroll
    for (int t = 0; t < 8; ++t) {
      v2f b = *(const v2f*)(bp + t * 32);   // one b64 load per fragment
      acc[t] = wmma_f32(a, b, acc[t]);
    }
  }

#pragma unroll
  for (int t = 0; t < 8; ++t) {
#pragma unroll
    for (int r = 0; r < 8; ++r) {
      int row = m0 + r + 8 * half;   // C layout: vgpr r, lanes<16 -> M=r, lanes>=16 -> M=8+r
      int col = t * 16 + lr;
      float v = acc[t][r];
      if (MODE == 1) v = fmaxf(v + bias[col], 0.0f);
      C[row * DF + col] = v;
    }
  }
}

// ---------------------------------------------------------------------------
// xs[i,:] = h[i,:] * dinv[i]^2 + b[:]   (self-loop term + bias, seeds scatter)
__global__ void selfloop_bias_k(const float* __restrict__ h, const float* __restrict__ dinv,
                                const float* __restrict__ b, float* __restrict__ out) {
  int i = blockIdx.x * blockDim.x + threadIdx.x;
  int node = i >> 5;
  if (node >= N_NODES) return;
  int lane = i & 31;
  float w = dinv[node];
  w *= w;
  float4 v = ((const float4*)(h + node * DF))[lane];
  float4 bb = ((const float4*)b)[lane];
  float4 o;
  o.x = v.x * w + bb.x;
  o.y = v.y * w + bb.y;
  o.z = v.z * w + bb.z;
  o.w = v.w * w + bb.w;
  ((float4*)(out + node * DF))[lane] = o;
}

// out[dst,:] += h[src,:] * dinv[src]*dinv[dst]   (one wave per edge, float4 gather)
__global__ void scatter_k(const float* __restrict__ h, const float* __restrict__ dinv,
                          const int* __restrict__ src, const int* __restrict__ dst,
                          float* __restrict__ out) {
  int i = blockIdx.x * blockDim.x + threadIdx.x;
  int e = i >> 5;
  if (e >= N_EDGES) return;
  int lane = i & 31;
  int s = src[e];
  int d = dst[e];
  float w = dinv[s] * dinv[d];
  float4 v = ((const float4*)(h + s * DF))[lane];
  float* op = out + d * DF + lane * 4;
  atomicAdd(op + 0, v.x * w);
  atomicAdd(op + 1, v.y * w);
  atomicAdd(op + 2, v.z * w);
  atomicAdd(op + 3, v.w * w);
}

// ---------------------------------------------------------------------------
// pooled[batch[m],:] += ([xs, xd, xs*xd, xs-xd][m,:] @ Wf[512x128]) + bf
// Wf passed in packed-pair layout. Virtual K=512 A matrix is built on the fly;
// epilogue scatters into pooled.
__global__ __launch_bounds__(256) void gemm_fuse_pool_k(
    const float* __restrict__ xs, const float* __restrict__ xd,
    const float* __restrict__ Wfp, const float* __restrict__ bf,
    const int* __restrict__ batch, float* __restrict__ pooled, int M) {
  int wave = (blockIdx.x * 256 + (int)threadIdx.x) >> 5;
  int lane = threadIdx.x & 31;
  int m0 = wave * 16;
  if (m0 >= M) return;
  int half = lane >> 4;
  int lr = lane & 15;
  const float* xsp = xs + (m0 + lr) * DF + 2 * half;
  const float* xdp = xd + (m0 + lr) * DF + 2 * half;
  v8f z = {};
  v8f acc[8];
#pragma unroll
  for (int t = 0; t < 8; ++t) acc[t] = z;

#pragma unroll
  for (int blk = 0; blk < 4; ++blk) {          // 0:xs 1:xd 2:xs*xd 3:xs-xd
    for (int k = 0; k < DF; k += 4) {
      v2f a;
      if (blk == 0) {
        a = *(const v2f*)(xsp + k);
      } else if (blk == 1) {
        a = *(const v2f*)(xdp + k);
      } else {
        v2f s = *(const v2f*)(xsp + k);
        v2f d = *(const v2f*)(xdp + k);
        a = (blk == 2) ? (v2f)(s * d) : (v2f)(s - d);
      }
      // global pair index: (blk*128 + k + 2*half)/2 = blk*64 + k/2 + half
      const float* bp = Wfp + ((blk * 64 + (k >> 1) + half) * DF + lr) * 2;
#pragma unroll
      for (int t = 0; t < 8; ++t) {
        v2f b = *(const v2f*)(bp + t * 32);
        acc[t] = wmma_f32(a, b, acc[t]);
      }
    }
  }

#pragma unroll
  for (int r = 0; r < 8; ++r) {
    int row = m0 + r + 8 * half;
    int g = batch[row];
    float* pp = pooled + g * DF;
#pragma unroll
    for (int t = 0; t < 8; ++t) {
      atomicAdd(pp + t * 16 + lr, acc[t][r] + bf[t * 16 + lr]);
    }
  }
}

// pooled /= max(counts, 1)
__global__ void pool_div_k(float* __restrict__ pooled, const float* __restrict__ counts) {
  int i = blockIdx.x * blockDim.x + threadIdx.x;
  if (i >= NG * DF) return;
  pooled[i] /= fmaxf(counts[i >> 7], 1.0f);
}

// out[g,c] = bh2[c] + hid[g,:] . Wh2[:,c]   (128 -> 2)
__global__ void head2_k(const float* __restrict__ hid, const float* __restrict__ W,
                        const float* __restrict__ b, float* __restrict__ out) {
  int i = blockIdx.x * blockDim.x + threadIdx.x;
  if (i >= NG * NC) return;
  int g = i / NC;
  int c = i % NC;
  float acc = b[c];
  const float* hp = hid + g * DF;
#pragma unroll 8
  for (int k = 0; k < DF; ++k) acc += hp[k] * W[k * NC + c];
  out[i] = acc;
}

// ---------------------------------------------------------------------------
extern "C" void kernel_launch(void* const* d_in, const int* in_sizes, int n_in,
                              void* d_out, int out_size, void* d_ws, size_t ws_size,
                              hipStream_t stream) {
  const float* x = (const float*)d_in[0];
  const int* ei = (const int*)d_in[1];
  const int* src = ei;              // edge_index[0]
  const int* dstp = ei + N_EDGES;   // edge_index[1]
  const int* batch = (const int*)d_in[2];
  const float* W1 = (const float*)d_in[3];
  const float* b1 = (const float*)d_in[4];
  const float* W2 = (const float*)d_in[5];
  const float* b2 = (const float*)d_in[6];
  const float* Wf = (const float*)d_in[7];
  const float* bf = (const float*)d_in[8];
  const float* Wh1 = (const float*)d_in[9];
  const float* bh1 = (const float*)d_in[10];
  const float* Wh2 = (const float*)d_in[11];
  const float* bh2 = (const float*)d_in[12];
  float* out = (float*)d_out;

  // workspace layout (floats), all offsets 16-float aligned
  float* ws = (float*)d_ws;
  float* dinv = ws;                               // 50048 (padded)
  float* bufH = dinv + 50048;                     // N*128 (h1 then h2)
  float* xs = bufH + (size_t)N_NODES * DF;        // N*128
  float* xd = xs + (size_t)N_NODES * DF;          // N*128
  float* pooled = xd + (size_t)N_NODES * DF;      // 512*128
  float* counts = pooled + NG * DF;               // 512
  float* hid = counts + NG;                       // 512*128
  float* pW1 = hid + NG * DF;                     // 64*128 float2 = 16384 f
  float* pW2 = pW1 + 64 * DF * 2;                 // 16384 f
  float* pWh1 = pW2 + 64 * DF * 2;                // 16384 f
  float* pWf = pWh1 + 64 * DF * 2;                // 256*128 float2 = 65536 f

  const int TB = 256;
  const int gemmBlocks = ((N_NODES / 16) * 32 + TB - 1) / TB;   // 3125 waves
  const int nodeVec = (N_NODES * 32 + TB - 1) / TB;
  const int edgeVec = (N_EDGES * 32 + TB - 1) / TB;

  // one-time weight repack into fragment-native pair layout
  pack_pairs_k<<<(64 * DF + TB - 1) / TB, TB, 0, stream>>>(W1, (float2*)pW1, 64);
  pack_pairs_k<<<(64 * DF + TB - 1) / TB, TB, 0, stream>>>(W2, (float2*)pW2, 64);
  pack_pairs_k<<<(64 * DF + TB - 1) / TB, TB, 0, stream>>>(Wh1, (float2*)pWh1, 64);
  pack_pairs_k<<<(256 * DF + TB - 1) / TB, TB, 0, stream>>>(Wf, (float2*)pWf, 256);

  // degrees / norm / graph sizes
  init_misc_k<<<(NG * DF + TB - 1) / TB, TB, 0, stream>>>(dinv, pooled, counts);
  deg_edges_k<<<(N_EDGES + TB - 1) / TB, TB, 0, stream>>>(dstp, dinv);
  dinv_counts_k<<<(N_NODES + TB - 1) / TB, TB, 0, stream>>>(dinv, batch, counts);

  // conv 1: h = x@W1 ; xs = scatter-norm(h) + self-loop + b1
  gemm128_k<0><<<gemmBlocks, TB, 0, stream>>>(x, pW1, nullptr, bufH, N_NODES);
  selfloop_bias_k<<<nodeVec, TB, 0, stream>>>(bufH, dinv, b1, xs);
  scatter_k<<<edgeVec, TB, 0, stream>>>(bufH, dinv, src, dstp, xs);

  // conv 2 (reuses bufH)
  gemm128_k<0><<<gemmBlocks, TB, 0, stream>>>(x, pW2, nullptr, bufH, N_NODES);
  selfloop_bias_k<<<nodeVec, TB, 0, stream>>>(bufH, dinv, b2, xd);
  scatter_k<<<edgeVec, TB, 0, stream>>>(bufH, dinv, src, dstp, xd);

  // interaction GEMM fused with segment-sum pooling
  gemm_fuse_pool_k<<<gemmBlocks, TB, 0, stream>>>(xs, xd, pWf, bf, batch, pooled, N_NODES);
  pool_div_k<<<(NG * DF + TB - 1) / TB, TB, 0, stream>>>(pooled, counts);

  // MLP head
  gemm128_k<1><<<((NG / 16) * 32 + TB - 1) / TB, TB, 0, stream>>>(pooled, pWh1, bh1, hid, NG);
  head2_k<<<(NG * NC + TB - 1) / TB, TB, 0, stream>>>(hid, Wh2, bh2, out);
}